// SpatialGather_Module_80092550135933
// MI455X (gfx1250) — compile-verified
//
#include <hip/hip_runtime.h>
#include <math.h>

typedef __attribute__((ext_vector_type(2))) float v2f;
typedef __attribute__((ext_vector_type(8))) float v8f;

#define BATCH   16
#define CCH     512
#define KCL     19
#define NSP     16384          // 128*128 spatial
#define CG      4              // c-tiles (of 16) per wave
#define CGROUPS (CCH / (16 * CG))  // 8
#define SEG     16             // n-segments
#define NSEG    (NSP / SEG)    // 1024

// ---------------------------------------------------------------------------
// Kernel 1: per-(b,k) softmax statistics.  shift = max + log(sum(exp(x-max)))
// so that p[n] = exp(x[n] - shift).  304 rows, one 256-thread block each.
// ---------------------------------------------------------------------------
__global__ void __launch_bounds__(256) sg_softmax_stats(
    const float* __restrict__ probs, float* __restrict__ shift) {
  const int pair = blockIdx.x;                 // b*19 + k, 0..303
  const float* row = probs + (size_t)pair * NSP;
  __shared__ float red[256];
  const int t = threadIdx.x;

  float mx = -INFINITY;
  for (int i = t; i < NSP; i += 256) mx = fmaxf(mx, row[i]);
  red[t] = mx; __syncthreads();
  for (int s = 128; s > 0; s >>= 1) {
    if (t < s) red[t] = fmaxf(red[t], red[t + s]);
    __syncthreads();
  }
  mx = red[0]; __syncthreads();

  float sum = 0.f;
  for (int i = t; i < NSP; i += 256) sum += __expf(row[i] - mx);
  red[t] = sum; __syncthreads();
  for (int s = 128; s > 0; s >>= 1) {
    if (t < s) red[t] += red[t + s];
    __syncthreads();
  }
  if (t == 0) shift[pair] = mx + __logf(red[0]);
}

// ---------------------------------------------------------------------------
// Kernel 2: zero the output (atomic accumulation target).
// ---------------------------------------------------------------------------
__global__ void __launch_bounds__(256) sg_zero_out(float* __restrict__ out, int n) {
  int i = blockIdx.x * 256 + threadIdx.x;
  if (i < n) out[i] = 0.f;
}

// ---------------------------------------------------------------------------
// Kernel 3: fused softmax x feats GEMM via V_WMMA_F32_16X16X4_F32.
//   D(16k x 16c) += A(16k x 4n) * B(4n x 16c)
// A lane layout: lane L, vgpr v -> M = L%16, K = 2*(L/16)+v  (one float2/lane)
// B lane layout: lane L, vgpr v -> N = L%16, K = 2*(L/16)+v  (one float2/lane)
// D lane layout: vgpr r, lane L -> M = r + 8*(L/16), N = L%16
// One wave per block; grid = B * CGROUPS * SEG, segments fastest for L2 reuse
// of the per-batch probs slice.
// ---------------------------------------------------------------------------
__global__ void __launch_bounds__(32) sg_gemm_wmma(
    const float* __restrict__ feats, const float* __restrict__ probs,
    const float* __restrict__ shift, float* __restrict__ out) {
  const int id = blockIdx.x;
  const int s  = id % SEG;
  const int cg = (id / SEG) % CGROUPS;
  const int b  = id / (SEG * CGROUPS);

  const int lane = threadIdx.x;
  const int lo = lane & 15;
  const int hi = lane >> 4;

  const int n0 = s * NSEG + 2 * hi;

  // A rows (k): tile0 rows 0..15 valid; tile1 rows 16..18, clamp address for
  // lanes whose row would run past k=18 (their D rows are never written).
  const int k0 = lo;
  int k1 = 16 + lo; if (k1 > KCL - 1) k1 = KCL - 1;
  const float* pA0 = probs + ((size_t)b * KCL + k0) * NSP + n0;
  const float* pA1 = probs + ((size_t)b * KCL + k1) * NSP + n0;
  const float sh0 = shift[b * KCL + k0];
  const float sh1 = shift[b * KCL + k1];

  // B columns (c): 4 tiles of 16 channels
  const int cbase = cg * (16 * CG);
  const float* pB0 = feats + ((size_t)b * CCH + cbase +  0 + lo) * NSP + n0;
  const float* pB1 = feats + ((size_t)b * CCH + cbase + 16 + lo) * NSP + n0;
  const float* pB2 = feats + ((size_t)b * CCH + cbase + 32 + lo) * NSP + n0;
  const float* pB3 = feats + ((size_t)b * CCH + cbase + 48 + lo) * NSP + n0;

  v8f acc[2][CG] = {};

  for (int n = 0; n < NSEG; n += 4) {
    const v2f ra0 = *(const v2f*)(pA0 + n);
    const v2f ra1 = *(const v2f*)(pA1 + n);
    v2f a0, a1;
    a0.x = __expf(ra0.x - sh0); a0.y = __expf(ra0.y - sh0);
    a1.x = __expf(ra1.x - sh1); a1.y = __expf(ra1.y - sh1);

    const v2f b0 = *(const v2f*)(pB0 + n);
    const v2f b1 = *(const v2f*)(pB1 + n);
    const v2f b2 = *(const v2f*)(pB2 + n);
    const v2f b3 = *(const v2f*)(pB3 + n);

    acc[0][0] = __builtin_amdgcn_wmma_f32_16x16x4_f32(false, a0, false, b0, (short)0, acc[0][0], false, false);
    acc[1][0] = __builtin_amdgcn_wmma_f32_16x16x4_f32(false, a1, false, b0, (short)0, acc[1][0], false, false);
    acc[0][1] = __builtin_amdgcn_wmma_f32_16x16x4_f32(false, a0, false, b1, (short)0, acc[0][1], false, false);
    acc[1][1] = __builtin_amdgcn_wmma_f32_16x16x4_f32(false, a1, false, b1, (short)0, acc[1][1], false, false);
    acc[0][2] = __builtin_amdgcn_wmma_f32_16x16x4_f32(false, a0, false, b2, (short)0, acc[0][2], false, false);
    acc[1][2] = __builtin_amdgcn_wmma_f32_16x16x4_f32(false, a1, false, b2, (short)0, acc[1][2], false, false);
    acc[0][3] = __builtin_amdgcn_wmma_f32_16x16x4_f32(false, a0, false, b3, (short)0, acc[0][3], false, false);
    acc[1][3] = __builtin_amdgcn_wmma_f32_16x16x4_f32(false, a1, false, b3, (short)0, acc[1][3], false, false);
  }

  // Writeback: out[b][c][k], k = t*16 + r + 8*hi, c = cbase + j*16 + lo
  #pragma unroll
  for (int t = 0; t < 2; ++t) {
    #pragma unroll
    for (int j = 0; j < CG; ++j) {
      #pragma unroll
      for (int r = 0; r < 8; ++r) {
        const int k = t * 16 + r + 8 * hi;
        if (k < KCL) {
          const int c = cbase + j * 16 + lo;
          unsafeAtomicAdd(&out[((size_t)b * CCH + c) * KCL + k], acc[t][j][r]);
        }
      }
    }
  }
}

// ---------------------------------------------------------------------------
extern "C" void kernel_launch(void* const* d_in, const int* in_sizes, int n_in,
                              void* d_out, int out_size, void* d_ws, size_t ws_size,
                              hipStream_t stream) {
  (void)in_sizes; (void)n_in; (void)ws_size;
  const float* feats = (const float*)d_in[0];   // [16, 512, 128, 128]
  const float* probs = (const float*)d_in[1];   // [16, 19, 128, 128]
  float* out   = (float*)d_out;                 // [16, 512, 19, 1]
  float* shift = (float*)d_ws;                  // 304 floats

  // 1) softmax stats per (b,k)
  sg_softmax_stats<<<BATCH * KCL, 256, 0, stream>>>(probs, shift);

  // 2) zero output
  sg_zero_out<<<(out_size + 255) / 256, 256, 0, stream>>>(out, out_size);

  // 3) fused softmax-GEMM with f32 WMMA
  sg_gemm_wmma<<<BATCH * CGROUPS * SEG, 32, 0, stream>>>(feats, probs, shift, out);
}